// DecoderWithAttention_618475290916
// MI455X (gfx1250) — compile-verified
//
#include <hip/hip_runtime.h>
#include <math.h>

typedef __attribute__((ext_vector_type(2))) float v2f;
typedef __attribute__((ext_vector_type(8))) float v8f;

#define B_   128
#define P_   196
#define ENC_ 2048
#define D_   1024
#define A_   1024
#define E_   1024
#define V_   32000
#define CAP_ 22
#define TM_  21
#define BD_  (B_*D_)

__device__ __forceinline__ float sigf(float x) { return 1.0f / (1.0f + expf(-x)); }

// gfx1250 async copy: global -> LDS, 16B per lane, tracked by ASYNCcnt.
__device__ __forceinline__ void async_b128(const float* gsrc, unsigned lds_off) {
  asm volatile("global_load_async_to_lds_b128 %0, %1, off"
               :: "v"(lds_off), "v"((unsigned long long)(uintptr_t)gsrc)
               : "memory");
}

// ------------------------------------------------------------------
// fp32 WMMA GEMM:  C[M,N] = act( A[M,K] @ W[N,K]^T + bias (+C) )
// Block tile 32(M) x 64(N), K chunk 32, double-buffered LDS staged with
// GLOBAL_LOAD_ASYNC_TO_LDS_B128. 4 waves (2x2); each wave owns two
// 16x16 fp32 accumulators fed by V_WMMA_F32_16X16X4_F32.
// All call sites use M%32==0, N%64==0, K%32==0 (no staging guards).
// ------------------------------------------------------------------
__global__ __launch_bounds__(128) void gemm_wmma_f32(
    const float* __restrict__ Ag, int lda, const int* __restrict__ rowmap,
    const float* __restrict__ W, int ldw,
    float* __restrict__ C, long ldc,
    const float* __restrict__ bias,
    int M, int N, int K, int addC, int act,
    const int* __restrict__ rowmask)
{
  // row stride 36 floats = 144B: keeps every b128 destination 16B aligned
  __shared__ float As[2][32][36];
  __shared__ float Ws[2][64][36];
  const int tid  = threadIdx.x;
  const int lane = tid & 31;
  const int wave = tid >> 5;
  const int wm = wave >> 1, wn = wave & 1;
  const int half = lane >> 4, l15 = lane & 15;
  const int mblk = blockIdx.y * 32;
  const int nblk = blockIdx.x * 64;

  v8f acc0 = {}; v8f acc1 = {};

  // issue async stage of K-chunk k0 into buffer `buf` (6 async ops per wave)
  auto stage = [&](int k0, int buf) {
    // A tile: 32 rows x 32 cols = 256 float4, 2 per thread
    #pragma unroll
    for (int i = tid; i < 32 * 8; i += 128) {
      int r = i >> 3, c4 = (i & 7) << 2;
      int pr = rowmap ? rowmap[mblk + r] : (mblk + r);
      async_b128(Ag + (long)pr * lda + k0 + c4,
                 (unsigned)(uintptr_t)&As[buf][r][c4]);
    }
    // W tile: 64 rows x 32 cols = 512 float4, 4 per thread
    #pragma unroll
    for (int i = tid; i < 64 * 8; i += 128) {
      int r = i >> 3, c4 = (i & 7) << 2;
      async_b128(W + (long)(nblk + r) * ldw + k0 + c4,
                 (unsigned)(uintptr_t)&Ws[buf][r][c4]);
    }
  };

  stage(0, 0);
  int buf = 0;
  for (int k0 = 0; k0 < K; k0 += 32) {
    if (k0 + 32 < K) {
      stage(k0 + 32, buf ^ 1);
      asm volatile("s_wait_asynccnt 6" ::: "memory");  // this chunk's 6 done
    } else {
      asm volatile("s_wait_asynccnt 0" ::: "memory");
    }
    __syncthreads();                                   // all waves' LDS ready
#pragma unroll
    for (int kk = 0; kk < 32; kk += 4) {
      const int kb = kk + half * 2;
      v2f a;  a.x  = As[buf][wm*16 + l15][kb];      a.y  = As[buf][wm*16 + l15][kb+1];
      v2f b0; b0.x = Ws[buf][wn*32 + l15][kb];      b0.y = Ws[buf][wn*32 + l15][kb+1];
      v2f b1; b1.x = Ws[buf][wn*32 + 16 + l15][kb]; b1.y = Ws[buf][wn*32 + 16 + l15][kb+1];
      acc0 = __builtin_amdgcn_wmma_f32_16x16x4_f32(false, a, false, b0, (short)0, acc0, false, false);
      acc1 = __builtin_amdgcn_wmma_f32_16x16x4_f32(false, a, false, b1, (short)0, acc1, false, false);
    }
    __syncthreads();                                   // done reading buf
    buf ^= 1;
  }

  const int m0 = mblk + wm*16;
  const int n0 = nblk + wn*32;
#pragma unroll
  for (int r = 0; r < 8; ++r) {
    int m = m0 + half*8 + r;
    if (m >= M) continue;
    int n1 = n0 + l15, n2 = n0 + 16 + l15;
    float v0 = acc0[r], v1 = acc1[r];
    if (n1 < N) {
      if (bias) v0 += bias[n1];
      if (addC) v0 += C[(long)m*ldc + n1];
      if (act == 1) v0 = sigf(v0); else if (act == 2) v0 = tanhf(v0);
      if (rowmask && !rowmask[m]) v0 = 0.f;
      C[(long)m*ldc + n1] = v0;
    }
    if (n2 < N) {
      if (bias) v1 += bias[n2];
      if (addC) v1 += C[(long)m*ldc + n2];
      if (act == 1) v1 = sigf(v1); else if (act == 2) v1 = tanhf(v1);
      if (rowmask && !rowmask[m]) v1 = 0.f;
      C[(long)m*ldc + n2] = v1;
    }
  }
}

// ---------------- sort by -lens (stable), emit caps/dec_len/sort_ind/rel_pos
__global__ void k_sort(const int* __restrict__ lens, const int* __restrict__ caps,
                       int* sort_ws, int* dec_ws, int* caps_ws,
                       float* out_caps, float* out_dlen, float* out_sind, float* out_rel)
{
  int i = threadIdx.x;
  if (i < B_) {
    int li = lens[i];
    int rank = 0;
    for (int j = 0; j < B_; ++j) {
      int lj = lens[j];
      if (lj > li || (lj == li && j < i)) rank++;
    }
    sort_ws[rank] = i;
  }
  __syncthreads();
  if (i < B_) {
    int s = sort_ws[i];
    int dl = lens[s] - 1;
    dec_ws[i] = dl;
    out_dlen[i] = (float)dl;
    out_sind[i] = (float)s;
    for (int j = 0; j < CAP_; ++j) {
      int cv = caps[s*CAP_ + j];
      caps_ws[i*CAP_ + j] = cv;
      out_caps[i*CAP_ + j] = (float)cv;
    }
    float dlf = (float)dl;
    for (int t = 0; t < TM_; ++t)
      out_rel[i*TM_ + t] = (t < dl) ? ((float)(t + 1)) / dlf : 0.f;
  }
}

__global__ void k_rowmap(const int* __restrict__ sort_ws, int* rowmap) {
  int i = blockIdx.x * blockDim.x + threadIdx.x;
  if (i < B_ * P_) rowmap[i] = sort_ws[i / P_] * P_ + (i % P_);
}

__global__ void k_active(const int* __restrict__ dec_ws, int* active) {
  int i = blockIdx.x * blockDim.x + threadIdx.x;
  if (i < TM_ * B_) active[i] = ((i / B_) < dec_ws[i % B_]) ? 1 : 0;
}

__global__ void k_zero(float* p, int n) {
  int i = blockIdx.x * blockDim.x + threadIdx.x;
  if (i < n) p[i] = 0.f;
}

__global__ void k_mean(const float* __restrict__ enc, const int* __restrict__ sort_ws,
                       float* mean_enc) {
  int i = blockIdx.x * blockDim.x + threadIdx.x;
  if (i >= B_ * ENC_) return;
  int b = i / ENC_, e = i % ENC_;
  long base = (long)sort_ws[b] * P_ * ENC_ + e;
  float s = 0.f;
  for (int p = 0; p < P_; ++p) s += enc[base + (long)p * ENC_];
  mean_enc[i] = s * (1.0f / (float)P_);
}

__global__ void k_embed(const float* __restrict__ emb, const int* __restrict__ caps_ws,
                        float* e_t, int t) {
  int i = blockIdx.x * blockDim.x + threadIdx.x;
  if (i >= B_ * E_) return;
  int b = i / E_, e = i % E_;
  e_t[i] = emb[(long)caps_ws[b*CAP_ + t] * E_ + e];
}

__global__ void k_lstm(const float* __restrict__ g, const float* __restrict__ c_old,
                       float* h_new, float* c_new) {
  int i = blockIdx.x * blockDim.x + threadIdx.x;
  if (i >= BD_) return;
  int b = i / D_, d = i % D_;
  const float* gr = g + (long)b * 4 * D_;
  float gi = gr[d], gf = gr[D_ + d], gg = gr[2*D_ + d], go = gr[3*D_ + d];
  float c2 = sigf(gf) * c_old[i] + sigf(gi) * tanhf(gg);
  h_new[i] = sigf(go) * tanhf(c2);
  c_new[i] = c2;
}

// scores[b,p] = sum_a tanh(att1[b,p,a] + att2[b,a]) * fW[a] + fb
__global__ __launch_bounds__(128) void k_scores(const float* __restrict__ att1,
                                                const float* __restrict__ att2,
                                                const float* __restrict__ fW,
                                                const float* __restrict__ fb,
                                                float* scores) {
  int bp = blockIdx.x * 4 + (threadIdx.x >> 5);
  if (bp >= B_ * P_) return;
  int b = bp / P_;
  int lane = threadIdx.x & 31;
  float s = 0.f;
  const float* a1 = att1 + (long)bp * A_;
  const float* a2 = att2 + (long)b * A_;
  for (int a = lane; a < A_; a += 32) s += tanhf(a1[a] + a2[a]) * fW[a];
  for (int off = 16; off > 0; off >>= 1) s += __shfl_xor(s, off);
  if (lane == 0) scores[bp] = s + fb[0];
}

__global__ __launch_bounds__(256) void k_softmax(const float* __restrict__ scores,
                                                 float* alpha) {
  __shared__ float red[256];
  int b = blockIdx.x, t = threadIdx.x;
  float v = (t < P_) ? scores[b*P_ + t] : -3.4e38f;
  red[t] = v; __syncthreads();
  for (int s = 128; s > 0; s >>= 1) { if (t < s) red[t] = fmaxf(red[t], red[t + s]); __syncthreads(); }
  float mx = red[0]; __syncthreads();
  float ex = (t < P_) ? expf(v - mx) : 0.f;
  red[t] = ex; __syncthreads();
  for (int s = 128; s > 0; s >>= 1) { if (t < s) red[t] += red[t + s]; __syncthreads(); }
  float inv = 1.0f / red[0];
  if (t < P_) alpha[b*P_ + t] = ex * inv;
}

// att[b,e] = gate[b,e] * sum_p alpha[b,p] * enc_sorted[b,p,e]
__global__ void k_awe_gate(const float* __restrict__ alpha, const float* __restrict__ enc,
                           const int* __restrict__ sort_ws, const float* __restrict__ gate,
                           float* attv) {
  int i = blockIdx.x * blockDim.x + threadIdx.x;
  if (i >= B_ * ENC_) return;
  int b = i / ENC_, e = i % ENC_;
  long base = (long)sort_ws[b] * P_ * ENC_ + e;
  const float* al = alpha + b * P_;
  float s = 0.f;
  for (int p = 0; p < P_; ++p) s += al[p] * enc[base + (long)p * ENC_];
  attv[i] = gate[i] * s;
}

__global__ __launch_bounds__(128) void k_rpm(const float* __restrict__ h2n,
                                             const float* __restrict__ rpmW,
                                             const int* __restrict__ active,
                                             float* out_pos, int t) {
  int b = blockIdx.x * 4 + (threadIdx.x >> 5);
  if (b >= B_) return;
  int lane = threadIdx.x & 31;
  float s = 0.f;
  for (int d = lane; d < D_; d += 32) s += h2n[(long)b*D_ + d] * rpmW[d];
  for (int off = 16; off > 0; off >>= 1) s += __shfl_xor(s, off);
  if (lane == 0) out_pos[b*TM_ + t] = active[t*B_ + b] ? sigf(s) : 0.f;
}

__global__ void k_update(float* h1, float* c1, float* h2, float* c2,
                         const float* h1n, const float* c1n,
                         const float* h2n, const float* c2n,
                         const int* __restrict__ active, int t) {
  int i = blockIdx.x * blockDim.x + threadIdx.x;
  if (i >= BD_) return;
  if (active[t*B_ + i / D_]) { h1[i] = h1n[i]; c1[i] = c1n[i]; h2[i] = h2n[i]; c2[i] = c2n[i]; }
}

// ------------------------------------------------------------------
static inline void gemm(hipStream_t st, const float* A, int lda, const int* rowmap,
                        const float* W, int ldw, float* C, long ldc, const float* bias,
                        int M, int N, int K, int addC, int act, const int* rowmask) {
  dim3 g((N + 63) / 64, (M + 31) / 32);
  gemm_wmma_f32<<<g, 128, 0, st>>>(A, lda, rowmap, W, ldw, C, ldc, bias, M, N, K, addC, act, rowmask);
}

extern "C" void kernel_launch(void* const* d_in, const int* in_sizes, int n_in,
                              void* d_out, int out_size, void* d_ws, size_t ws_size,
                              hipStream_t stream) {
  (void)in_sizes; (void)n_in; (void)out_size; (void)ws_size;
  const float* enc      = (const float*)d_in[0];
  const int*   caps     = (const int*)  d_in[1];
  const int*   lens     = (const int*)  d_in[2];
  const float* emb      = (const float*)d_in[3];
  const float* encAttW  = (const float*)d_in[4];
  const float* encAttB  = (const float*)d_in[5];
  const float* decAttW  = (const float*)d_in[6];
  const float* decAttB  = (const float*)d_in[7];
  const float* fullAttW = (const float*)d_in[8];
  const float* fullAttB = (const float*)d_in[9];
  const float* fBetaW   = (const float*)d_in[10];
  const float* fBetaB   = (const float*)d_in[11];
  const float* l1Wih    = (const float*)d_in[12];
  const float* l1Whh    = (const float*)d_in[13];
  const float* l1bih    = (const float*)d_in[14];
  const float* l1bhh    = (const float*)d_in[15];
  const float* l2Wih    = (const float*)d_in[16];
  const float* l2Whh    = (const float*)d_in[17];
  const float* l2bih    = (const float*)d_in[18];
  const float* l2bhh    = (const float*)d_in[19];
  const float* ramW     = (const float*)d_in[20];
  const float* ramB     = (const float*)d_in[21];
  const float* rpmW     = (const float*)d_in[22];

  // ---- output layout (flat concat, all float)
  float* out      = (float*)d_out;
  float* out_pred = out;                                  // B*TM*V
  float* out_caps = out_pred + (long)B_ * TM_ * V_;       // B*CAP
  float* out_dlen = out_caps + B_ * CAP_;                 // B
  float* out_sind = out_dlen + B_;                        // B
  float* out_rel  = out_sind + B_;                        // B*TM
  float* out_pos  = out_rel  + B_ * TM_;                  // B*TM

  // ---- workspace layout
  char* p = (char*)d_ws;
  auto alloc = [&](size_t bytes) { char* q = p; p += (bytes + 255) & ~size_t(255); return q; };
  int*   sort_ws = (int*)  alloc(B_ * 4);
  int*   dec_ws  = (int*)  alloc(B_ * 4);
  int*   caps_ws = (int*)  alloc(B_ * CAP_ * 4);
  int*   rowmap  = (int*)  alloc((size_t)B_ * P_ * 4);
  int*   active  = (int*)  alloc((size_t)TM_ * B_ * 4);
  float* att1    = (float*)alloc((size_t)B_ * P_ * A_ * 4);
  float* gbuf    = (float*)alloc((size_t)B_ * 4 * D_ * 4);
  float* hc      = (float*)alloc((size_t)4 * BD_ * 4);    // h1,c1,h2,c2 contiguous
  float* h1 = hc, *c1 = hc + BD_, *h2 = hc + 2*BD_, *c2 = hc + 3*BD_;
  float* h1n     = (float*)alloc((size_t)BD_ * 4);
  float* c1n     = (float*)alloc((size_t)BD_ * 4);
  float* h2n     = (float*)alloc((size_t)BD_ * 4);
  float* c2n     = (float*)alloc((size_t)BD_ * 4);
  float* e_t     = (float*)alloc((size_t)B_ * E_ * 4);
  float* meanE   = (float*)alloc((size_t)B_ * ENC_ * 4);
  float* att2    = (float*)alloc((size_t)B_ * A_ * 4);
  float* scores  = (float*)alloc((size_t)B_ * P_ * 4);
  float* alpha   = (float*)alloc((size_t)B_ * P_ * 4);
  float* gate    = (float*)alloc((size_t)B_ * ENC_ * 4);
  float* attv    = (float*)alloc((size_t)B_ * ENC_ * 4);

  // ---- setup
  k_sort<<<1, 128, 0, stream>>>(lens, caps, sort_ws, dec_ws, caps_ws,
                                out_caps, out_dlen, out_sind, out_rel);
  k_rowmap<<<(B_*P_ + 255)/256, 256, 0, stream>>>(sort_ws, rowmap);
  k_active<<<(TM_*B_ + 255)/256, 256, 0, stream>>>(dec_ws, active);
  k_zero<<<(4*BD_ + 255)/256, 256, 0, stream>>>(hc, 4*BD_);
  k_mean<<<(B_*ENC_ + 255)/256, 256, 0, stream>>>(enc, sort_ws, meanE);

  // att1 = enc_sorted @ enc_att_W.T + b   (M=B*P, N=A, K=ENC) — once
  gemm(stream, enc, ENC_, rowmap, encAttW, ENC_, att1, A_, encAttB,
       B_*P_, A_, ENC_, 0, 0, nullptr);

  for (int t = 0; t < TM_; ++t) {
    k_embed<<<(B_*E_ + 255)/256, 256, 0, stream>>>(emb, caps_ws, e_t, t);

    // g1 = [e_t, h2, mean_enc] @ l1_Wih.T + bih + h1 @ l1_Whh.T + bhh (K-sliced)
    gemm(stream, e_t,  E_,   nullptr, l1Wih,            E_+D_+ENC_, gbuf, 4*D_, l1bih, B_, 4*D_, E_,   0, 0, nullptr);
    gemm(stream, h2,   D_,   nullptr, l1Wih + E_,       E_+D_+ENC_, gbuf, 4*D_, nullptr, B_, 4*D_, D_,  1, 0, nullptr);
    gemm(stream, meanE,ENC_, nullptr, l1Wih + E_ + D_,  E_+D_+ENC_, gbuf, 4*D_, nullptr, B_, 4*D_, ENC_,1, 0, nullptr);
    gemm(stream, h1,   D_,   nullptr, l1Whh,            D_,         gbuf, 4*D_, l1bhh, B_, 4*D_, D_,   1, 0, nullptr);
    k_lstm<<<(BD_ + 255)/256, 256, 0, stream>>>(gbuf, c1, h1n, c1n);

    // att2 = h1n @ dec_att_W.T + b
    gemm(stream, h1n, D_, nullptr, decAttW, D_, att2, A_, decAttB, B_, A_, D_, 0, 0, nullptr);
    k_scores<<<(B_*P_ + 3)/4, 128, 0, stream>>>(att1, att2, fullAttW, fullAttB, scores);
    k_softmax<<<B_, 256, 0, stream>>>(scores, alpha);

    // gate = sigmoid(h1n @ f_beta_W.T + b) ; att = gate * (alpha . enc)
    gemm(stream, h1n, D_, nullptr, fBetaW, D_, gate, ENC_, fBetaB, B_, ENC_, D_, 0, 1, nullptr);
    k_awe_gate<<<(B_*ENC_ + 255)/256, 256, 0, stream>>>(alpha, enc, sort_ws, gate, attv);

    // g2 = [att, h1n] @ l2_Wih.T + bih2 + h2 @ l2_Whh.T + bhh2
    gemm(stream, attv, ENC_, nullptr, l2Wih,        ENC_+D_, gbuf, 4*D_, l2bih, B_, 4*D_, ENC_, 0, 0, nullptr);
    gemm(stream, h1n,  D_,   nullptr, l2Wih + ENC_, ENC_+D_, gbuf, 4*D_, nullptr, B_, 4*D_, D_,  1, 0, nullptr);
    gemm(stream, h2,   D_,   nullptr, l2Whh,        D_,      gbuf, 4*D_, l2bhh, B_, 4*D_, D_,   1, 0, nullptr);
    k_lstm<<<(BD_ + 255)/256, 256, 0, stream>>>(gbuf, c2, h2n, c2n);

    // preds (masked) -> out[:, t, :]   (ldc = TM*V, row offset t*V)
    gemm(stream, h2n, D_, nullptr, ramW, D_, out_pred + (long)t * V_, (long)TM_ * V_,
         ramB, B_, V_, D_, 0, 0, active + t * B_);

    k_rpm<<<(B_ + 3)/4, 128, 0, stream>>>(h2n, rpmW, active, out_pos, t);
    k_update<<<(BD_ + 255)/256, 256, 0, stream>>>(h1, c1, h2, c2, h1n, c1n, h2n, c2n, active, t);
  }
}